// GCN_1_21337397526794
// MI455X (gfx1250) — compile-verified
//
#include <hip/hip_runtime.h>
#include <hip/hip_bf16.h>

typedef float v2f __attribute__((ext_vector_type(2)));
typedef float v8f __attribute__((ext_vector_type(8)));

// ---------------------------------------------------------------------------
// Utility kernels
// ---------------------------------------------------------------------------
__global__ void fill_f32(float* __restrict__ p, float v, long n) {
  long i = (long)blockIdx.x * blockDim.x + threadIdx.x;
  if (i < n) p[i] = v;
}

__global__ void deg_accum(const long long* __restrict__ dst,
                          const float* __restrict__ w,
                          float* __restrict__ deg, int E) {
  int e = blockIdx.x * blockDim.x + threadIdx.x;
  if (e >= E) return;
  atomicAdd(&deg[(int)dst[e]], w[e]);
}

__global__ void rsqrt_deg(const float* __restrict__ deg,
                          float* __restrict__ dinv, int n) {
  int i = blockIdx.x * blockDim.x + threadIdx.x;
  if (i >= n) return;
  float d = deg[i];
  dinv[i] = (d > 0.f) ? rsqrtf(d) : 0.f;
}

// ---------------------------------------------------------------------------
// Dense linear transform  H[N,OUTC] = X[N,INC] @ W[OUTC,INC]^T
// via V_WMMA_F32_16X16X4_F32.  One wave per 16-row M-tile; W staged in LDS.
// A-frag lane layout (16x4 f32): row M = lane&15; K = k0 + (lane>=16?2:0) + v.
// B-frag (4x16):                 col N = lane&15; K = k0 + (lane>=16?2:0) + v.
// C/D (16x16 f32, 8 VGPRs):      col N = lane&15; row M = (lane>=16?8:0) + v.
// ---------------------------------------------------------------------------
template <int INC, int OUTC>
__global__ void gemm_xwT_wmma(const float* __restrict__ X,
                              const float* __restrict__ W,   // [OUTC][INC]
                              float* __restrict__ H,
                              int nrows) {
  __shared__ float sW[OUTC * INC];
  for (int i = threadIdx.x; i < OUTC * INC; i += blockDim.x) sW[i] = W[i];
  __syncthreads();

  const int lane    = threadIdx.x & 31;
  const int wave    = threadIdx.x >> 5;
  const int mtile   = blockIdx.x * (blockDim.x >> 5) + wave;
  const int m0      = mtile * 16;
  if (m0 >= nrows) return;          // wave-uniform exit (EXEC all-1 for WMMA)

  const int half    = lane >> 4;    // 0 or 1
  const int koff    = half * 2;
  const int sub     = lane & 15;
  const float* xrow = X + (size_t)(m0 + sub) * INC;

  constexpr int NT = OUTC / 16;
  v8f acc[NT] = {};

#pragma unroll
  for (int k0 = 0; k0 < INC; k0 += 4) {
    v2f a;
    a.x = xrow[k0 + koff];
    a.y = xrow[k0 + koff + 1];
#pragma unroll
    for (int t = 0; t < NT; ++t) {
      const int n = t * 16 + sub;
      v2f b;
      b.x = sW[n * INC + k0 + koff];
      b.y = sW[n * INC + k0 + koff + 1];
      acc[t] = __builtin_amdgcn_wmma_f32_16x16x4_f32(
          /*neg_a=*/false, a, /*neg_b=*/false, b,
          /*c_mod=*/(short)0, acc[t], /*reuse_a=*/false, /*reuse_b=*/false);
    }
  }

#pragma unroll
  for (int t = 0; t < NT; ++t) {
    const int n = t * 16 + sub;
#pragma unroll
    for (int v = 0; v < 8; ++v) {
      const int m = half * 8 + v;
      H[(size_t)(m0 + m) * OUTC + n] = acc[t][v];
    }
  }
}

// ---------------------------------------------------------------------------
// Edge-parallel message scatter: agg[dst] += h[src] * (dinv[s]*w*dinv[d]).
// h (<=25.6MB) fits in the 192MB L2, so gathers and float atomics stay in L2.
// ---------------------------------------------------------------------------
template <int C>
__global__ void scatter_edges(const long long* __restrict__ src,
                              const long long* __restrict__ dst,
                              const float* __restrict__ w,
                              const float* __restrict__ dinv,
                              const float* __restrict__ h,
                              float* __restrict__ agg, int E) {
  int e = blockIdx.x * blockDim.x + threadIdx.x;
  if (e >= E) return;
  const int s = (int)src[e];
  const int d = (int)dst[e];
  const float nrm = dinv[s] * w[e] * dinv[d];
  const float* hs = h + (size_t)s * C;
  float* ad = agg + (size_t)d * C;
  __builtin_prefetch(hs, 0, 0);   // global_prefetch_b8 on the gathered row
#pragma unroll
  for (int f = 0; f < C; ++f) atomicAdd(&ad[f], hs[f] * nrm);
}

// ---------------------------------------------------------------------------
// Epilogue: self-loop term + bias (+ optional ReLU).
// out[i,f] = agg[i,f] + h[i,f]*dinv[i]^2 + bias[f]
// ---------------------------------------------------------------------------
template <int C, bool RELU>
__global__ void finish_nodes(const float* __restrict__ agg,
                             const float* __restrict__ h,
                             const float* __restrict__ dinv,
                             const float* __restrict__ bias,
                             float* __restrict__ out, int nrows) {
  long i = (long)blockIdx.x * blockDim.x + threadIdx.x;
  long node = i / C;
  int  f    = (int)(i % C);
  if (node >= nrows) return;
  const float di = dinv[node];
  float v = agg[i] + h[i] * di * di + bias[f];
  if (RELU) v = fmaxf(v, 0.f);
  out[i] = v;
}

// ---------------------------------------------------------------------------
// Launcher
// ---------------------------------------------------------------------------
extern "C" void kernel_launch(void* const* d_in, const int* in_sizes, int n_in,
                              void* d_out, int out_size, void* d_ws, size_t ws_size,
                              hipStream_t stream) {
  const float*     x   = (const float*)d_in[0];
  const long long* ei  = (const long long*)d_in[1];  // [2, E] int64
  const float*     ew  = (const float*)d_in[2];
  const float*     W1  = (const float*)d_in[3];      // [64, 64]
  const float*     b1  = (const float*)d_in[4];      // [64]
  const float*     W2  = (const float*)d_in[5];      // [32, 64]
  const float*     b2  = (const float*)d_in[6];      // [32]
  float*           out = (float*)d_out;

  constexpr int INC = 64, HID = 64, OC = 32;
  const int E = in_sizes[2];
  const int N = in_sizes[0] / INC;                   // 100000 (multiple of 16)

  const long long* src = ei;
  const long long* dst = ei + E;

  float* ws   = (float*)d_ws;
  float* deg  = ws;                                  // N
  float* dinv = deg  + N;                            // N
  float* bufA = dinv + N;                            // N*64
  float* bufB = bufA + (size_t)N * HID;              // N*64
  float* bufC = bufB + (size_t)N * HID;              // N*32

  const int TB = 256;
  auto blk = [](long n, int tb) { return (int)((n + tb - 1) / tb); };
  const int mtiles      = N / 16;
  const int gemm_blocks = (mtiles + (TB / 32) - 1) / (TB / 32);

  // --- normalization: deg = 1 (self-loop) + sum_w at dst; dinv = rsqrt(deg)
  fill_f32<<<blk(N, TB), TB, 0, stream>>>(deg, 1.0f, N);
  deg_accum<<<blk(E, TB), TB, 0, stream>>>(dst, ew, deg, E);
  rsqrt_deg<<<blk(N, TB), TB, 0, stream>>>(deg, dinv, N);

  // --- layer 1: h = x @ W1^T  (WMMA f32)
  gemm_xwT_wmma<INC, HID><<<gemm_blocks, TB, 0, stream>>>(x, W1, bufA, N);
  fill_f32<<<blk((long)N * HID, TB), TB, 0, stream>>>(bufB, 0.0f, (long)N * HID);
  scatter_edges<HID><<<blk(E, TB), TB, 0, stream>>>(src, dst, ew, dinv, bufA, bufB, E);
  finish_nodes<HID, true><<<blk((long)N * HID, TB), TB, 0, stream>>>(
      bufB, bufA, dinv, b1, bufB, N);                // in-place -> h1 in bufB

  // --- layer 2: h2 = h1 @ W2^T  (WMMA f32)
  gemm_xwT_wmma<HID, OC><<<gemm_blocks, TB, 0, stream>>>(bufB, W2, bufC, N);
  fill_f32<<<blk((long)N * OC, TB), TB, 0, stream>>>(bufA, 0.0f, (long)N * OC);
  scatter_edges<OC><<<blk(E, TB), TB, 0, stream>>>(src, dst, ew, dinv, bufC, bufA, E);
  finish_nodes<OC, false><<<blk((long)N * OC, TB), TB, 0, stream>>>(
      bufA, bufC, dinv, b2, out, N);
}